// SigKernelRidgeRegression_3917010174816
// MI455X (gfx1250) — compile-verified
//
#include <hip/hip_runtime.h>
#include <hip/hip_bf16.h>

typedef __attribute__((ext_vector_type(2))) float v2f;
typedef __attribute__((ext_vector_type(8))) float v8f;

#define NB   24      // batch
#define L0   48      // raw length
#define D0   6       // raw channels
#define LR   189     // refined length (47*4+1)
#define LP   192     // padded to 12 WMMA tiles
#define DK   8       // feature dim padded (time + 6 + zero)
#define LM   188     // LR-1 (inc dims, PDE L)
#define GST  190     // G row stride in words: PDE lane stride 189, gcd(189,64)=1 -> conflict-free
#define NPAIR 576    // 24*24

// One workgroup per (n,m) path pair. g=0: K(train,train), g=1: Kc(test,train).
// Fully fused: refine -> fp32 WMMA dot matrix -> exp -> PDE wavefront, all in LDS.
__global__ __launch_bounds__(256)
void sig_gram_kernel(const float* __restrict__ Xtr,
                     const float* __restrict__ Xte,
                     float* __restrict__ Kout /* [2*576] */)
{
    __shared__ float Xs[LP][DK];     // refined row path (A side), col 7 = 0
    __shared__ float Ys[LP][DK];     // refined col path (B side)
    __shared__ float Gs[LR * GST];   // RBF kernel matrix 189x189 (stride 190)
    __shared__ float xs2[LP];
    __shared__ float ys2[LP];
    __shared__ float wf[3][LP];      // PDE wavefront: prev2/prev1/cur (rotated)

    const int pair = blockIdx.x;
    const int g    = pair / NPAIR;
    const int nm   = pair % NPAIR;
    const int n    = nm / NB;
    const int m    = nm % NB;
    const float* __restrict__ Xrow = (g ? Xte : Xtr) + (size_t)n * (L0 * D0);
    const float* __restrict__ Yrow = Xtr + (size_t)m * (L0 * D0);

    const int tid = threadIdx.x;

    // ---- add-time + refine (factor 4): s_j = 0.25*j, exact in fp32 ----
    if (tid < LP) {
        const int j = tid;
        if (j < LR) {
            int i0 = j >> 2;
            if (i0 > L0 - 2) i0 = L0 - 2;
            const float frac = 0.25f * (float)j - (float)i0;
            const float omf  = 1.0f - frac;
            const float t0 = (float)i0 * (1.0f / 47.0f);
            const float t1 = (float)(i0 + 1) * (1.0f / 47.0f);
            const float tj = omf * t0 + frac * t1;
            Xs[j][0] = tj;
            Ys[j][0] = tj;
#pragma unroll
            for (int d = 0; d < D0; ++d) {
                Xs[j][1 + d] = omf * Xrow[i0 * D0 + d] + frac * Xrow[(i0 + 1) * D0 + d];
                Ys[j][1 + d] = omf * Yrow[i0 * D0 + d] + frac * Yrow[(i0 + 1) * D0 + d];
            }
            Xs[j][7] = 0.0f;
            Ys[j][7] = 0.0f;
        } else {
#pragma unroll
            for (int k = 0; k < DK; ++k) { Xs[j][k] = 0.0f; Ys[j][k] = 0.0f; }
        }
    }
    __syncthreads();

    if (tid < LP) {
        float sx = 0.0f, sy = 0.0f;
#pragma unroll
        for (int k = 0; k < 7; ++k) {
            sx += Xs[tid][k] * Xs[tid][k];
            sy += Ys[tid][k] * Ys[tid][k];
        }
        xs2[tid] = sx;
        ys2[tid] = sy;
    }
    __syncthreads();

    // ---- G = exp(-0.5*(|x|^2+|y|^2-2 x.y)) via V_WMMA_F32_16X16X4_F32 ----
    // A (16x4 f32): lanes 0-15 hold M=0..15; VGPR0 = K0(lo)/K2(hi), VGPR1 = K1(lo)/K3(hi)
    // B (4x16 f32): one K-row per half-wave per VGPR (mirror of A with M<->N)
    const int wave   = tid >> 5;   // 8 waves
    const int lane   = tid & 31;
    const int mloc   = lane & 15;
    const int hi     = lane >> 4;  // 0/1
    const int ka     = hi ? 2 : 0;

    for (int t = wave; t < 144; t += 8) {
        const int tm = t / 12;
        const int tn = t % 12;
        const int arow = tm * 16 + mloc;
        const int bcol = tn * 16 + mloc;
        v2f a0, a1, b0, b1;
        a0.x = Xs[arow][ka];     a0.y = Xs[arow][ka + 1];
        a1.x = Xs[arow][ka + 4]; a1.y = Xs[arow][ka + 5];
        b0.x = Ys[bcol][ka];     b0.y = Ys[bcol][ka + 1];
        b1.x = Ys[bcol][ka + 4]; b1.y = Ys[bcol][ka + 5];
        v8f c = {};
        c = __builtin_amdgcn_wmma_f32_16x16x4_f32(false, a0, false, b0, (short)0, c, false, false);
        c = __builtin_amdgcn_wmma_f32_16x16x4_f32(false, a1, false, b1, (short)0, c, false, false);

        // D layout: element e -> (M = hi*8 + e, N = mloc)
        const int rbase = tm * 16 + hi * 8;
        const int cc    = tn * 16 + mloc;

        if (tm < 11 && tn < 11) {
            // interior tile: rows <= 175, cols <= 175 -> no guards, straight-line
            const float y2 = ys2[cc];
#pragma unroll
            for (int e = 0; e < 8; ++e) {
                const float sq = xs2[rbase + e] + y2 - 2.0f * c[e];
                Gs[(rbase + e) * GST + cc] = __expf(-0.5f * sq);
            }
        } else {
            // boundary tile: compute everything unguarded, guard only the store
            const float y2   = ys2[cc];
            const bool colok = (cc < LR);
            float gv[8];
#pragma unroll
            for (int e = 0; e < 8; ++e) {
                const float sq = xs2[rbase + e] + y2 - 2.0f * c[e];
                gv[e] = __expf(-0.5f * sq);
            }
#pragma unroll
            for (int e = 0; e < 8; ++e) {
                if (colok && (rbase + e) < LR)
                    Gs[(rbase + e) * GST + cc] = gv[e];
            }
        }
    }

    // ---- PDE wavefront: 375 anti-diagonal steps, inc = 2nd difference of G ----
    if (tid < LP) { wf[0][tid] = 1.0f; wf[1][tid] = 1.0f; }
    __syncthreads();

    int ip2 = 0, ip1 = 1, ic = 2;
    for (int d = 2; d <= 2 * LM; ++d) {
        if (tid < LR) {
            const int ii = tid;
            int i_idx = ii - 1;
            if (i_idx < 0) i_idx = 0;
            if (i_idx > LM - 1) i_idx = LM - 1;
            int j_idx = d - ii - 1;
            if (j_idx < 0) j_idx = 0;
            if (j_idx > LM - 1) j_idx = LM - 1;
            const float g00 = Gs[i_idx * GST + j_idx];
            const float g01 = Gs[i_idx * GST + j_idx + 1];
            const float g10 = Gs[(i_idx + 1) * GST + j_idx];
            const float g11 = Gs[(i_idx + 1) * GST + j_idx + 1];
            const float inc  = g11 - g10 - g01 + g00;
            const float i212 = inc * inc * (1.0f / 12.0f);
            const float a = 1.0f + 0.5f * inc + i212;
            const float b = 1.0f - i212;
            const float p1  = wf[ip1][ii];
            const float p1m = (ii > 0) ? wf[ip1][ii - 1] : 0.0f;
            const float p2m = (ii > 0) ? wf[ip2][ii - 1] : 0.0f;
            float nv = (p1m + p1) * a - p2m * b;
            if (ii == 0 || ii == d) nv = 1.0f;
            wf[ic][ii] = nv;
        }
        __syncthreads();
        const int tmp = ip2; ip2 = ip1; ip1 = ic; ic = tmp;
    }

    if (tid == 0) Kout[pair] = wf[ip1][LM];
}

// Single-block: alphas = (K + I)^-1 Y  (Gauss-Jordan, K+I is SPD), out = Kc @ alphas
__global__ __launch_bounds__(256)
void sig_solve_kernel(const float* __restrict__ Kall /* [2*576] */,
                      const float* __restrict__ Ytr  /* [24*8] */,
                      float* __restrict__ out        /* [24*8] */)
{
    __shared__ float Ms[NB][NB + 1];
    __shared__ float Yl[NB][8];
    __shared__ float fcol[NB];
    __shared__ float s_inv;

    const int tid = threadIdx.x;

    for (int i = tid; i < NB * NB; i += 256) {
        const int r = i / NB, c = i % NB;
        Ms[r][c] = Kall[i] + ((r == c) ? 1.0f : 0.0f);   // REG = 1.0
    }
    if (tid < NB * 8) Yl[tid >> 3][tid & 7] = Ytr[tid];
    __syncthreads();

    for (int k = 0; k < NB; ++k) {
        if (tid == 0) s_inv = 1.0f / Ms[k][k];
        __syncthreads();
        if (tid < NB)                      Ms[k][tid] *= s_inv;
        else if (tid >= 32 && tid < 40)    Yl[k][tid - 32] *= s_inv;
        else if (tid >= 64 && tid < 64+NB) {
            const int r = tid - 64;
            fcol[r] = (r == k) ? 0.0f : Ms[r][k];
        }
        __syncthreads();
        for (int idx = tid; idx < NB * 32; idx += 256) {
            const int r = idx >> 5, j = idx & 31;
            if (r == k) continue;
            const float f = fcol[r];
            if (j < NB)       Ms[r][j] -= f * Ms[k][j];
            else if (j < NB + 8) Yl[r][j - NB] -= f * Yl[k][j - NB];
        }
        __syncthreads();
    }

    if (tid < NB * 8) {
        const int nn = tid >> 3, y = tid & 7;
        const float* __restrict__ Kc = Kall + NPAIR;
        float acc = 0.0f;
#pragma unroll
        for (int mm = 0; mm < NB; ++mm) acc += Kc[nn * NB + mm] * Yl[mm][y];
        out[tid] = acc;   // (24,1,8) flat
    }
}

extern "C" void kernel_launch(void* const* d_in, const int* in_sizes, int n_in,
                              void* d_out, int out_size, void* d_ws, size_t ws_size,
                              hipStream_t stream) {
    const float* Xtr = (const float*)d_in[0];   // (24,48,6)
    const float* Ytr = (const float*)d_in[1];   // (24,8)
    const float* Xte = (const float*)d_in[2];   // (24,48,6)
    float* Kall = (float*)d_ws;                 // [0..575]=K, [576..1151]=Kc

    sig_gram_kernel<<<dim3(2 * NPAIR), dim3(256), 0, stream>>>(Xtr, Xte, Kall);
    sig_solve_kernel<<<dim3(1), dim3(256), 0, stream>>>(Kall, Ytr, (float*)d_out);
}